// KFPooling_15255723835785
// MI455X (gfx1250) — compile-verified
//
#include <hip/hip_runtime.h>
#include <math.h>

#define EPS_F 1e-7f

typedef __attribute__((ext_vector_type(2))) float v2f;
typedef __attribute__((ext_vector_type(4))) float v4f;
typedef __attribute__((ext_vector_type(8))) float v8f;

// Problem constants (from setup_inputs)
static constexpr int Nq   = 1024;   // queries
static constexpr int Mm   = 2048;   // memories
static constexpr int DMEM = 512;
static constexpr int DH   = 128;
static constexpr int DP   = 144;    // hyperboloid dim 129 padded to multiple of 16

__device__ __forceinline__ float wave_sum(float v) {
#pragma unroll
  for (int o = 16; o > 0; o >>= 1) v += __shfl_xor(v, o, 32);
  return v;
}

// ---------------------------------------------------------------------------
// fp32 WMMA NT-GEMM:  C[M x N] = A[M x K] * B^T,  B row-major [N x K].
// K is a compile-time constant -> straight-line prefetch prologue, no in-loop
// guards, and a software-pipelined (double-buffered) k-loop: fragments for
// step k+4 are in flight while the NTILE V_WMMA_F32_16X16X4_F32 of step k
// issue, so waits become partial instead of load-to-use stalls.
// One wave owns a 16 x (16*NTILE) strip (A fragment reused NTILE times).
// Block = 128 threads = 4 waves; wave w owns row-tile 4*blockIdx.x + w.
// A layout: lane(l=lane&15) = M-row; v2f = K {2h, 2h+1}, h = lane>>4.
// B layout: lane = N-col;            v2f = K {2h, 2h+1}.
// C layout: VGPR r -> row (r + 8h), col = l.
// ---------------------------------------------------------------------------
template <int NTILE, int K>
__global__ void wmma_gemm_nt(const float* __restrict__ A, int lda,
                             const float* __restrict__ B, int ldb,
                             float* __restrict__ C, int ldc) {
  const int wave = threadIdx.x >> 5;
  const int lane = threadIdx.x & 31;
  const int h    = lane >> 4;
  const int l    = lane & 15;
  const int tm   = (blockIdx.x * 4 + wave) * 16;
  const int tn0  = blockIdx.y * 16 * NTILE;

  const float* arow = A + (size_t)(tm + l) * lda + 2 * h;
  const float* brow[NTILE];
#pragma unroll
  for (int t = 0; t < NTILE; ++t)
    brow[t] = B + (size_t)(tn0 + t * 16 + l) * ldb + 2 * h;

  // Straight-line K-panel prefetch (emits global_prefetch_b8, no branches)
#pragma unroll
  for (int p = 0; p < K; p += 64) {
    __builtin_prefetch(arow + p, 0, 1);
#pragma unroll
    for (int t = 0; t < NTILE; ++t) __builtin_prefetch(brow[t] + p, 0, 1);
  }

  v8f acc[NTILE];
#pragma unroll
  for (int t = 0; t < NTILE; ++t) acc[t] = (v8f){};

  // Pipeline prologue: fragments for k = 0
  v2f a_cur = *(const v2f*)(arow);
  v2f b_cur[NTILE];
#pragma unroll
  for (int t = 0; t < NTILE; ++t) b_cur[t] = *(const v2f*)(brow[t]);

#pragma unroll 4
  for (int k = 0; k < K - 4; k += 4) {
    // Issue next-step loads before consuming current fragments
    v2f a_nxt = *(const v2f*)(arow + k + 4);
    v2f b_nxt[NTILE];
#pragma unroll
    for (int t = 0; t < NTILE; ++t) b_nxt[t] = *(const v2f*)(brow[t] + k + 4);
#pragma unroll
    for (int t = 0; t < NTILE; ++t)
      acc[t] = __builtin_amdgcn_wmma_f32_16x16x4_f32(false, a_cur, false, b_cur[t],
                                                     (short)0, acc[t], false, false);
    a_cur = a_nxt;
#pragma unroll
    for (int t = 0; t < NTILE; ++t) b_cur[t] = b_nxt[t];
  }
  // Pipeline epilogue: last k-step
#pragma unroll
  for (int t = 0; t < NTILE; ++t)
    acc[t] = __builtin_amdgcn_wmma_f32_16x16x4_f32(false, a_cur, false, b_cur[t],
                                                   (short)0, acc[t], false, false);

#pragma unroll
  for (int t = 0; t < NTILE; ++t)
#pragma unroll
    for (int r = 0; r < 8; ++r)
      C[(size_t)(tm + r + 8 * h) * ldc + tn0 + t * 16 + l] = acc[t][r];
}

// ---------------------------------------------------------------------------
// Simple transpose: out[C x R] = in[R x C]^T  (one-time, tiny weight matrices)
// ---------------------------------------------------------------------------
__global__ void transpose_kernel(const float* __restrict__ in,
                                 float* __restrict__ out, int R, int C) {
  int idx = blockIdx.x * 256 + threadIdx.x;
  int r = idx / C, c = idx % C;
  out[(size_t)c * R + r] = in[idx];
}

// ---------------------------------------------------------------------------
// Hyperbolic lift: row x (128) -> expmap0(to_tangent0(x))  (dim 129, pad 144)
// H[0]=cosh(t), H[1+d]=sinh(t)*factor*x_d/t, t=max(||x||*factor, eps),
// factor = min(3.5/(||x||+eps), 1).  Hneg: time component negated (turns
// Lorentz inner products into plain WMMA dot products).  HT: transposed copy
// (DP x rows) so the pooled-tangent GEMM is also an NT GEMM.
// 8 rows per 256-thread block, one wave32 per row.
// ---------------------------------------------------------------------------
__global__ void hyp_map_kernel(const float* __restrict__ X,
                               float* __restrict__ H,
                               float* __restrict__ Hneg,
                               float* __restrict__ HT, int ldT) {
  const int row  = blockIdx.x * 8 + (threadIdx.x >> 5);
  const int lane = threadIdx.x & 31;
  v4f x = *(const v4f*)(X + (size_t)row * DH + lane * 4);
  float ss = x.x * x.x + x.y * x.y + x.z * x.z + x.w * x.w;
  ss = wave_sum(ss);
  float n      = sqrtf(ss);
  float factor = fminf(3.5f / (n + EPS_F), 1.0f);
  float t      = fmaxf(n * factor, EPS_F);
  float h0     = coshf(t);
  float coef   = sinhf(t) * factor / t;
  float y0 = coef * x.x, y1 = coef * x.y, y2 = coef * x.z, y3 = coef * x.w;

  float* hrow = H + (size_t)row * DP;
  if (lane == 0) hrow[0] = h0;
  hrow[1 + lane * 4 + 0] = y0;
  hrow[1 + lane * 4 + 1] = y1;
  hrow[1 + lane * 4 + 2] = y2;
  hrow[1 + lane * 4 + 3] = y3;
  if (lane < 15) hrow[129 + lane] = 0.0f;

  if (Hneg) {
    float* grow = Hneg + (size_t)row * DP;
    if (lane == 0) grow[0] = -h0;
    grow[1 + lane * 4 + 0] = y0;
    grow[1 + lane * 4 + 1] = y1;
    grow[1 + lane * 4 + 2] = y2;
    grow[1 + lane * 4 + 3] = y3;
    if (lane < 15) grow[129 + lane] = 0.0f;
  }
  if (HT) {
    if (lane == 0) HT[row] = h0;
    HT[(size_t)(1 + lane * 4 + 0) * ldT + row] = y0;
    HT[(size_t)(1 + lane * 4 + 1) * ldT + row] = y1;
    HT[(size_t)(1 + lane * 4 + 2) * ldT + row] = y2;
    HT[(size_t)(1 + lane * 4 + 3) * ldT + row] = y3;
    if (lane < 15) HT[(size_t)(129 + lane) * ldT + row] = 0.0f;
  }
}

// ---------------------------------------------------------------------------
// Fused softmax + closed-form logmap weights.  Per query row i:
//   t_j   = -BETA * sim_ij ; alpha = softmax(t)
//   dist  = acosh(max(-xy, 1+eps)) ; denom = sqrt(max(xy^2-1, eps))
//           (on the k=1 hyperboloid <nomin,nomin>_L == xy^2 - 1 exactly)
//   W_ij  = alpha*dist/denom   (overwrites sim in place)
//   s_i   = sum_j W_ij * xy_ij
// The two 8KB rows are staged to LDS with the CDNA5 async-to-LDS engine
// (GLOBAL_LOAD_ASYNC_TO_LDS_B128, ASYNCcnt) instead of VGPR loads.
// NOTE: static __shared__ starts at LDS offset 0 for this kernel.
// ---------------------------------------------------------------------------
__global__ void softmax_combine_kernel(float* __restrict__ simW,
                                       const float* __restrict__ xy,
                                       float* __restrict__ s_out) {
  const float BETA = 0.08838834764831845f;   // 1/sqrt(128)
  const int row = blockIdx.x;
  const int t   = threadIdx.x;               // 256 threads, 8 vals each
  float* rowp       = simW + (size_t)row * Mm;
  const float* xrow = xy   + (size_t)row * Mm;

  __shared__ float smem[2 * Mm];             // [0,2048)=sim row, [2048,4096)=xy row
  __shared__ float red[256];

  // Async-copy both rows into LDS: 4 x (256 lanes x 16B) transfers.
#pragma unroll
  for (int i = 0; i < 2; ++i) {
    unsigned ldsoff = (unsigned)(i * 4096 + t * 16);
    const float* g  = rowp + i * 1024 + t * 4;
    asm volatile("global_load_async_to_lds_b128 %0, %1, off"
                 :: "v"(ldsoff), "v"(g) : "memory");
  }
#pragma unroll
  for (int i = 0; i < 2; ++i) {
    unsigned ldsoff = (unsigned)(8192 + i * 4096 + t * 16);
    const float* g  = xrow + i * 1024 + t * 4;
    asm volatile("global_load_async_to_lds_b128 %0, %1, off"
                 :: "v"(ldsoff), "v"(g) : "memory");
  }
  asm volatile("s_wait_asynccnt 0x0" ::: "memory");
  __syncthreads();

  float v[8];
  float m = -INFINITY;
#pragma unroll
  for (int i = 0; i < 8; ++i) {
    v[i] = -BETA * smem[t + i * 256];
    m = fmaxf(m, v[i]);
  }
  red[t] = m; __syncthreads();
  for (int o = 128; o > 0; o >>= 1) { if (t < o) red[t] = fmaxf(red[t], red[t + o]); __syncthreads(); }
  m = red[0]; __syncthreads();

  float sum = 0.f;
#pragma unroll
  for (int i = 0; i < 8; ++i) { v[i] = expf(v[i] - m); sum += v[i]; }
  red[t] = sum; __syncthreads();
  for (int o = 128; o > 0; o >>= 1) { if (t < o) red[t] += red[t + o]; __syncthreads(); }
  const float inv = 1.0f / red[0]; __syncthreads();

  float acc = 0.f;
#pragma unroll
  for (int i = 0; i < 8; ++i) {
    float x    = smem[Mm + t + i * 256];
    float dist = acoshf(fmaxf(-x, 1.0f + EPS_F));
    float den  = sqrtf(fmaxf(x * x - 1.0f, EPS_F));
    float w    = v[i] * inv * dist / den;
    rowp[t + i * 256] = w;
    acc += w * x;
  }
  red[t] = acc; __syncthreads();
  for (int o = 128; o > 0; o >>= 1) { if (t < o) red[t] += red[t + o]; __syncthreads(); }
  if (t == 0) s_out[row] = red[0];
}

// ---------------------------------------------------------------------------
// Finalize: u = T_i + s_i*Q_i ; z = expmap(Q,u) ; out = logmap0(z)[1:]
// One wave32 per row (8 rows per block).
// ---------------------------------------------------------------------------
__global__ void finalize_kernel(const float* __restrict__ T,
                                const float* __restrict__ Qh,
                                const float* __restrict__ S,
                                float* __restrict__ out) {
  const int row  = blockIdx.x * 8 + (threadIdx.x >> 5);
  const int lane = threadIdx.x & 31;
  const float si = S[row];
  const float* trow = T  + (size_t)row * DP;
  const float* qrow = Qh + (size_t)row * DP;

  float u[5], q[5], z[5];
  float uu = 0.f;
  int c = 0;
  for (int d = lane; d < DP; d += 32, ++c) {
    float qq = qrow[d];
    float uv = trow[d] + si * qq;
    u[c] = uv; q[c] = qq;
    uu += (d == 0) ? -uv * uv : uv * uv;      // Lorentz <u,u>
  }
  uu = wave_sum(uu);
  float un = sqrtf(fmaxf(uu, EPS_F));
  float ch = coshf(un);
  float sh = sinhf(un) / un;

  float zn = 0.f;
  c = 0;
  for (int d = lane; d < DP; d += 32, ++c) {
    float zz = ch * q[c] + sh * u[c];
    z[c] = zz;
    if (d != 0) zn += zz * zz;
  }
  zn = wave_sum(zn);
  float z0    = __shfl(z[0], 0, 32);          // lane 0 slot 0 holds d==0
  float dist  = acoshf(fmaxf(z0, 1.0f + EPS_F));
  float nn    = fmaxf(sqrtf(zn), EPS_F);
  float scale = dist / nn;

  float* orow = out + (size_t)row * DH;
  c = 0;
  for (int d = lane; d < DP; d += 32, ++c)
    if (d >= 1 && d <= DH) orow[d - 1] = scale * z[c];
}

// ---------------------------------------------------------------------------
extern "C" void kernel_launch(void* const* d_in, const int* in_sizes, int n_in,
                              void* d_out, int out_size, void* d_ws, size_t ws_size,
                              hipStream_t stream) {
  (void)in_sizes; (void)n_in; (void)out_size; (void)ws_size;
  const float* queries = (const float*)d_in[0];   // 1024 x 128
  const float* keys    = (const float*)d_in[1];   // 2048 x 512
  const float* values  = (const float*)d_in[2];   // 2048 x 512
  const float* W_K     = (const float*)d_in[3];   // 512 x 128
  const float* W_V     = (const float*)d_in[4];   // 128 x 128
  float* out = (float*)d_out;                     // 1024 x 128

  char* ws = (char*)d_ws;
  auto alloc = [&](size_t bytes) {
    float* p = (float*)ws;
    ws += (bytes + 255) & ~(size_t)255;
    return p;
  };
  float* WKT = alloc((size_t)DH * DMEM * 4); // W_K^T  (128 x 512)
  float* WVT = alloc((size_t)DH * DH * 4);   // W_V^T  (128 x 128)
  float* KW  = alloc((size_t)Mm * DH * 4);   // keys@W_K
  float* VW  = alloc((size_t)Mm * DH * 4);   // values@W_K
  float* VW2 = alloc((size_t)Mm * DH * 4);   // (values@W_K)@W_V
  float* Qh  = alloc((size_t)Nq * DP * 4);   // lifted queries
  float* Qn  = alloc((size_t)Nq * DP * 4);   // lifted queries, time negated
  float* Kh  = alloc((size_t)Mm * DP * 4);
  float* Vh  = alloc((size_t)Mm * DP * 4);
  float* VhT = alloc((size_t)DP * Mm * 4);   // Vh^T (144 x 2048)
  float* SW  = alloc((size_t)Nq * Mm * 4);   // sim -> logmap-weight matrix (in place)
  float* XY  = alloc((size_t)Nq * Mm * 4);   // Lorentz <Q,V>
  float* Tm  = alloc((size_t)Nq * DP * 4);   // W @ Vh
  float* Sv  = alloc((size_t)Nq * 4);        // rank-1 coefficients

  const dim3 gblk(128);  // 4 waves / block

  // One-time weight transposes so every GEMM is NT (contiguous b64 fragments)
  transpose_kernel<<<(DMEM * DH) / 256, 256, 0, stream>>>(W_K, WKT, DMEM, DH);
  transpose_kernel<<<(DH * DH) / 256,  256, 0, stream>>>(W_V, WVT, DH, DH);

  // Projections (fp32 WMMA, NT, 4-wide N strips)
  wmma_gemm_nt<4, DMEM><<<dim3(Mm/64, DH/64), gblk, 0, stream>>>(keys,   DMEM, WKT, DMEM, KW,  DH);
  wmma_gemm_nt<4, DMEM><<<dim3(Mm/64, DH/64), gblk, 0, stream>>>(values, DMEM, WKT, DMEM, VW,  DH);
  wmma_gemm_nt<4, DH>  <<<dim3(Mm/64, DH/64), gblk, 0, stream>>>(VW,     DH,   WVT, DH,   VW2, DH);

  // Hyperbolic lifts (values also emit Vh^T for the pooled GEMM)
  hyp_map_kernel<<<Nq/8, 256, 0, stream>>>(queries, Qh, Qn,      nullptr, 0);
  hyp_map_kernel<<<Mm/8, 256, 0, stream>>>(KW,  Kh, nullptr,     nullptr, 0);
  hyp_map_kernel<<<Mm/8, 256, 0, stream>>>(VW2, Vh, nullptr,     VhT,     Mm);

  // Lorentz score GEMMs: sim = Qn @ Kh^T, xy = Qn @ Vh^T  (pads are zero)
  wmma_gemm_nt<4, DP><<<dim3(Nq/64, Mm/64), gblk, 0, stream>>>(Qn, DP, Kh, DP, SW, Mm);
  wmma_gemm_nt<4, DP><<<dim3(Nq/64, Mm/64), gblk, 0, stream>>>(Qn, DP, Vh, DP, XY, Mm);

  // Softmax fused with closed-form logmap weights (async-to-LDS staged)
  softmax_combine_kernel<<<Nq, 256, 0, stream>>>(SW, XY, Sv);

  // Pooled tangent: T = W @ (VhT)^T   (the 1024x2048x129 logmap tensor, collapsed)
  wmma_gemm_nt<3, Mm><<<dim3(Nq/64, DP/48), gblk, 0, stream>>>(SW, Mm, VhT, Mm, Tm, DP);

  // expmap + logmap0
  finalize_kernel<<<Nq/8, 256, 0, stream>>>(Tm, Qh, Sv, out);
}